// Attention_21552145891583
// MI455X (gfx1250) — compile-verified
//
#include <hip/hip_runtime.h>

// Multi-head attention forward for MI455X (gfx1250, wave32).
// fp32 end-to-end via V_WMMA_F32_16X16X4_F32 (problem is HBM-bound at 23.3 TB/s,
// ~65 GFLOP vs ~250MB traffic, so fp32 WMMA keeps reference accuracy at no cost).
// K/A tiles are DMA'd into padded LDS rows by the Tensor Data Mover
// (tensor_load_to_lds + s_wait_tensorcnt); B/V tiles are transposed through
// VGPRs so every WMMA fragment is a contiguous 8-byte ds_load_b64.

#define DIMC   768
#define NHEAD  12
#define HD     64
#define BATCH  8
#define SEQ    1024
#define ASCALE 0.125f   // HD^-0.5

typedef __attribute__((ext_vector_type(2))) float v2f;
typedef __attribute__((ext_vector_type(8))) float v8f;
typedef __attribute__((ext_vector_type(4))) unsigned int v4u;
typedef __attribute__((ext_vector_type(8))) int v8i;
typedef __attribute__((ext_vector_type(4))) int v4i;

#if defined(__gfx1250__) && __has_builtin(__builtin_amdgcn_tensor_load_to_lds)
#define HAS_TDM 1
#else
#define HAS_TDM 0
#endif

__device__ __forceinline__ v8f wmma4(v2f a, v2f b, v8f c) {
  // D = A(16x4) * B(4x16) + C(16x16), fp32
  return __builtin_amdgcn_wmma_f32_16x16x4_f32(
      /*neg_a=*/false, a, /*neg_b=*/false, b,
      /*c_mod=*/(short)0, c, /*reuse_a=*/false, /*reuse_b=*/false);
}

#if HAS_TDM
// Low 32 bits of a flat pointer into the LDS aperture = workgroup LDS byte offset.
__device__ __forceinline__ unsigned int lds_off_of(const void* p) {
  return (unsigned int)(uintptr_t)p;
}

// TDM 2-D tile load: tile_w x tile_h elements (f32) from a row-major tensor with
// row stride `row_stride` into LDS at lds_off, inserting pad_amt_code+1 DWORDs of
// padding every 2^(pad_int_code+1) DWORDs (-> padded LDS row stride).
__device__ __forceinline__ void tdm_load_2d(unsigned int lds_off, const float* gptr,
                                            unsigned int tile_w, unsigned int tile_h,
                                            unsigned int tens_w, unsigned int tens_h,
                                            unsigned int row_stride,
                                            unsigned int pad_int_code,
                                            unsigned int pad_amt_code)
{
  unsigned long long ga = (unsigned long long)(uintptr_t)gptr;
  v4u g0;
  g0[0] = 1u;                                   // count=1 (valid user descriptor)
  g0[1] = lds_off;                              // lds_addr (bytes)
  g0[2] = (unsigned int)ga;                     // global_addr[31:0]
  g0[3] = (unsigned int)(ga >> 32) | 0x80000000u;  // global_addr[56:32] | type=2
  v8i g1;
  g1[0] = (int)((2u << 16)                      // data_size = 4 bytes
              | (1u << 20)                      // pad_enable
              | (pad_int_code << 22) | (pad_amt_code << 25));
  g1[1] = (int)((tens_w & 0xFFFFu) << 16);                       // tensor_dim0 lo
  g1[2] = (int)((tens_w >> 16) | ((tens_h & 0xFFFFu) << 16));    // dim0 hi, dim1 lo
  g1[3] = (int)((tens_h >> 16) | (tile_w << 16));                // dim1 hi, tile_dim0
  g1[4] = (int)tile_h;                                           // tile_dim1 (dim2=0)
  g1[5] = (int)row_stride;                                       // dim0_stride[31:0]
  g1[6] = 0;
  g1[7] = 0;
  v4i gz = {0, 0, 0, 0};
  v8i gx = {0, 0, 0, 0, 0, 0, 0, 0};
  // clang-23 (amdgpu-toolchain) 6-arg form: groups 0-3, extra group, cache policy
  __builtin_amdgcn_tensor_load_to_lds(g0, g1, gz, gz, gx, 0);
}
#endif

// ---------------------------------------------------------------------------
// Generic 64x64-tile GEMM, 4 waves / 128 threads per block.
// mode 0: C = A[8192,768] @ Bw[768,2304], scatter into Q/K/V [B,H,N,D]
// mode 1: C = A[8192,768] @ Bw[768,768] + bias, write to out [B*N, 768]
// ---------------------------------------------------------------------------
__global__ __launch_bounds__(128)
void gemm64x64_kernel(const float* __restrict__ A, const float* __restrict__ Bw,
                      int ldb, int mode,
                      float* __restrict__ qo, float* __restrict__ ko,
                      float* __restrict__ vo,
                      const float* __restrict__ bias, float* __restrict__ out)
{
  __shared__ float As[64 * 20];   // 64x16 A tile, padded row stride 20
  __shared__ float Bt[64 * 18];   // 16x64 B tile TRANSPOSED: Bt[col][k], stride 18

  const int tid  = threadIdx.x;
  const int w    = tid >> 5;      // wave id 0..3 -> rows 16*w..16*w+15
  const int lane = tid & 31;
  const int lm   = lane & 15;
  const int hi   = lane >> 4;
  const int rowBase = blockIdx.y * 64;
  const int colBase = blockIdx.x * 64;

  v8f acc[4];
  #pragma unroll
  for (int j = 0; j < 4; ++j)
    #pragma unroll
    for (int r = 0; r < 8; ++r) acc[j][r] = 0.0f;

  for (int k0 = 0; k0 < DIMC; k0 += 16) {
    __syncthreads();
#if HAS_TDM
    // each wave DMAs its own 16x16 A slice; pad 16 DWORDs -> +4 => stride 20
    tdm_load_2d(lds_off_of(&As[w * 16 * 20]),
                &A[(size_t)(rowBase + w * 16) * DIMC + k0],
                /*tile_w=*/16, /*tile_h=*/16,
                /*tens_w=*/DIMC, /*tens_h=*/BATCH * SEQ,
                /*row_stride=*/DIMC, /*pad_int=*/3, /*pad_amt=*/3);
#else
    #pragma unroll
    for (int i = 0; i < 2; ++i) {
      int slot = tid + i * 128;
      int row = slot >> 2, c4 = (slot & 3) << 2;
      float4 t = *(const float4*)&A[(size_t)(rowBase + row) * DIMC + k0 + c4];
      *(float4*)&As[row * 20 + c4] = t;
    }
#endif
    // cooperative load: B tile 16x64, store transposed into Bt[col][k]
    // (overlaps the in-flight TDM transfer)
    #pragma unroll
    for (int i = 0; i < 2; ++i) {
      int slot = tid + i * 128;
      int kr = slot >> 4, c4 = (slot & 15) << 2;
      float4 t = *(const float4*)&Bw[(size_t)(k0 + kr) * ldb + colBase + c4];
      Bt[(c4 + 0) * 18 + kr] = t.x;
      Bt[(c4 + 1) * 18 + kr] = t.y;
      Bt[(c4 + 2) * 18 + kr] = t.z;
      Bt[(c4 + 3) * 18 + kr] = t.w;
      if (k0 + 16 < DIMC)
        __builtin_prefetch(&Bw[(size_t)(k0 + 16 + kr) * ldb + colBase + c4], 0, 3);
    }
#if HAS_TDM
    __builtin_amdgcn_s_wait_tensorcnt(0);
#endif
    __syncthreads();

    #pragma unroll
    for (int kk = 0; kk < 4; ++kk) {               // 4 WMMA k-steps of K=4
      // A frag: A(m,k) -> vgpr=k%2, lane=m+16*(k/2)  => contiguous float2
      v2f a = *(const v2f*)&As[(16 * w + lm) * 20 + kk * 4 + 2 * hi];
      #pragma unroll
      for (int j = 0; j < 4; ++j) {                // 4 n-subtiles
        // B frag: B(k,n) -> vgpr=k%2, lane=n+16*(k/2) => contiguous in Bt
        v2f b = *(const v2f*)&Bt[(j * 16 + lm) * 18 + kk * 4 + 2 * hi];
        acc[j] = wmma4(a, b, acc[j]);
      }
    }
  }

  // epilogue: C layout -> vgpr r, lanes0-15: M=r, lanes16-31: M=r+8
  #pragma unroll
  for (int j = 0; j < 4; ++j) {
    int col = colBase + j * 16 + lm;
    #pragma unroll
    for (int r = 0; r < 8; ++r) {
      int row = rowBase + 16 * w + r + 8 * hi;
      float val = acc[j][r];
      if (mode == 0) {
        int s  = col / DIMC, rem = col % DIMC;
        int h  = rem / HD,   dd  = rem % HD;
        int b  = row / SEQ,  n   = row % SEQ;
        size_t idx = (((size_t)(b * NHEAD + h)) * SEQ + n) * HD + dd;
        (s == 0 ? qo : (s == 1 ? ko : vo))[idx] = val;
      } else {
        out[(size_t)row * DIMC + col] = val + bias[col];
      }
    }
  }
}

// ---------------------------------------------------------------------------
// Flash attention: block = (b, h, 64 query rows); 4 waves, each owns 16 rows.
// Q frags in registers; K (TDM, padded rows) and V (transposed) in LDS;
// S=Q.K^T and O+=P.V via fp32 WMMA; online softmax in C-fragment layout with
// shfl_xor row reductions; P converted C->A layout through per-wave LDS.
// ---------------------------------------------------------------------------
__global__ __launch_bounds__(128)
void attn_kernel(const float* __restrict__ Q, const float* __restrict__ K,
                 const float* __restrict__ V, float* __restrict__ Out)
{
  __shared__ float Ks[64 * 68];       // [key][d], padded row stride 68
  __shared__ float Vt[64 * 18];       // TRANSPOSED: Vt[d][key], stride 18
  __shared__ float Ps[4 * 16 * 68];   // per-wave P staging (C-layout -> A-layout)

  const int tid  = threadIdx.x;
  const int w    = tid >> 5;
  const int lane = tid & 31;
  const int lm   = lane & 15;
  const int hi   = lane >> 4;
  const int rblk = blockIdx.x & 15;     // 1024/64 = 16 query blocks
  const int bh   = blockIdx.x >> 4;     // 0..95

  const float* Qg = Q + (size_t)bh * SEQ * HD + (size_t)(rblk * 64 + w * 16) * HD;
  const float* Kg = K + (size_t)bh * SEQ * HD;
  const float* Vg = V + (size_t)bh * SEQ * HD;

  // Q tile 16x64 held as 16 A-fragments (float2 each)
  v2f qf[16];
  #pragma unroll
  for (int kk = 0; kk < 16; ++kk)
    qf[kk] = *(const v2f*)&Qg[lm * HD + kk * 4 + 2 * hi];

  v8f O[4];
  float m_r[8], l_r[8], alpha[8];
  #pragma unroll
  for (int j = 0; j < 4; ++j)
    #pragma unroll
    for (int r = 0; r < 8; ++r) O[j][r] = 0.0f;
  #pragma unroll
  for (int r = 0; r < 8; ++r) { m_r[r] = -1e30f; l_r[r] = 0.0f; }

  float* Pw = Ps + w * (16 * 68);

  for (int kb = 0; kb < 16; ++kb) {
    __syncthreads();   // prior iteration done reading Ks/Vt
#if HAS_TDM
    // each wave DMAs a 64x16 K slice; pad every 64 DWORDs by 4 => stride 68
    tdm_load_2d(lds_off_of(&Ks[w * 16 * 68]),
                &Kg[(size_t)(kb * 64 + w * 16) * HD],
                /*tile_w=*/HD, /*tile_h=*/16,
                /*tens_w=*/HD, /*tens_h=*/SEQ,
                /*row_stride=*/HD, /*pad_int=*/5, /*pad_amt=*/3);
#endif
    // manual transposed V copy overlaps the in-flight K DMA
    #pragma unroll
    for (int i = 0; i < 8; ++i) {      // 1024 float4 slots / 128 threads
      int slot = tid + i * 128;
      int row = slot >> 4, c4 = (slot & 15) << 2;
#if !HAS_TDM
      *(float4*)&Ks[row * 68 + c4] =
          *(const float4*)&Kg[(size_t)(kb * 64 + row) * HD + c4];
#endif
      float4 tv = *(const float4*)&Vg[(size_t)(kb * 64 + row) * HD + c4];
      Vt[(c4 + 0) * 18 + row] = tv.x;
      Vt[(c4 + 1) * 18 + row] = tv.y;
      Vt[(c4 + 2) * 18 + row] = tv.z;
      Vt[(c4 + 3) * 18 + row] = tv.w;
      if (kb + 1 < 16) {               // prefetch next key block
        __builtin_prefetch(&Kg[(size_t)((kb + 1) * 64 + row) * HD + c4], 0, 3);
        __builtin_prefetch(&Vg[(size_t)((kb + 1) * 64 + row) * HD + c4], 0, 3);
      }
    }
#if HAS_TDM
    __builtin_amdgcn_s_wait_tensorcnt(0);
#endif
    __syncthreads();

    // S = Q . K^T for 16 rows x 64 keys (4 tiles, 16 k-steps over D=64)
    v8f S[4];
    #pragma unroll
    for (int j = 0; j < 4; ++j)
      #pragma unroll
      for (int r = 0; r < 8; ++r) S[j][r] = 0.0f;

    #pragma unroll
    for (int kk = 0; kk < 16; ++kk) {
      v2f a = qf[kk];
      #pragma unroll
      for (int j = 0; j < 4; ++j) {
        // B(d,key) = K[key][d] -> contiguous float2 per lane
        v2f b = *(const v2f*)&Ks[(j * 16 + lm) * 68 + kk * 4 + 2 * hi];
        S[j] = wmma4(a, b, S[j]);
      }
    }

    // online softmax; row r (lanes 0-15) / row r+8 (lanes 16-31) per vgpr r
    #pragma unroll
    for (int r = 0; r < 8; ++r) {
      float s0 = S[0][r] * ASCALE, s1 = S[1][r] * ASCALE;
      float s2 = S[2][r] * ASCALE, s3 = S[3][r] * ASCALE;
      float mx = fmaxf(fmaxf(s0, s1), fmaxf(s2, s3));
      mx = fmaxf(mx, __shfl_xor(mx, 1, 32));
      mx = fmaxf(mx, __shfl_xor(mx, 2, 32));
      mx = fmaxf(mx, __shfl_xor(mx, 4, 32));
      mx = fmaxf(mx, __shfl_xor(mx, 8, 32));
      float mnew = fmaxf(m_r[r], mx);
      float al = __expf(m_r[r] - mnew);
      m_r[r] = mnew;
      alpha[r] = al;
      float p0 = __expf(s0 - mnew), p1 = __expf(s1 - mnew);
      float p2 = __expf(s2 - mnew), p3 = __expf(s3 - mnew);
      S[0][r] = p0; S[1][r] = p1; S[2][r] = p2; S[3][r] = p3;
      float rs = (p0 + p1) + (p2 + p3);
      rs += __shfl_xor(rs, 1, 32);
      rs += __shfl_xor(rs, 2, 32);
      rs += __shfl_xor(rs, 4, 32);
      rs += __shfl_xor(rs, 8, 32);
      l_r[r] = l_r[r] * al + rs;
    }
    #pragma unroll
    for (int j = 0; j < 4; ++j)
      #pragma unroll
      for (int r = 0; r < 8; ++r) O[j][r] *= alpha[r];

    // stage P through LDS: C layout -> rows of [16][64(+4)] (wave-private,
    // DS ops are in-order within a wave, so no barrier needed)
    #pragma unroll
    for (int j = 0; j < 4; ++j)
      #pragma unroll
      for (int r = 0; r < 8; ++r)
        Pw[(r + 8 * hi) * 68 + j * 16 + lm] = S[j][r];

    // O += P . V  (key dim 64 = 16 k-steps, 4 d-subtiles)
    #pragma unroll
    for (int kk = 0; kk < 16; ++kk) {
      v2f a = *(const v2f*)&Pw[lm * 68 + kk * 4 + 2 * hi];
      #pragma unroll
      for (int dj = 0; dj < 4; ++dj) {
        // B(key,d) = V[key][d] = Vt[d][key] -> contiguous float2 per lane
        v2f b = *(const v2f*)&Vt[(dj * 16 + lm) * 18 + kk * 4 + 2 * hi];
        O[dj] = wmma4(a, b, O[dj]);
      }
    }
  }

  // normalize and store as [B, N, H*D] so the proj gemm reads rows directly
  const int b = bh / NHEAD, h = bh % NHEAD;
  #pragma unroll
  for (int r = 0; r < 8; ++r) {
    float inv = 1.0f / l_r[r];
    int qrow = rblk * 64 + w * 16 + r + 8 * hi;
    size_t base = ((size_t)(b * SEQ + qrow)) * DIMC + h * HD;
    #pragma unroll
    for (int dj = 0; dj < 4; ++dj)
      Out[base + dj * 16 + lm] = O[dj][r] * inv;
  }
}

// ---------------------------------------------------------------------------
extern "C" void kernel_launch(void* const* d_in, const int* in_sizes, int n_in,
                              void* d_out, int out_size, void* d_ws, size_t ws_size,
                              hipStream_t stream) {
  const float* x      = (const float*)d_in[0];   // [8,1024,768]
  const float* w_qkv  = (const float*)d_in[1];   // [768, 2304]
  const float* w_proj = (const float*)d_in[2];   // [768, 768]
  const float* b_proj = (const float*)d_in[3];   // [768]
  float* out = (float*)d_out;                    // [8,1024,768]

  // workspace: Q, K, V in [B,H,N,D] + attn output in [B,N,768] (~100.7 MB)
  const size_t per = (size_t)BATCH * NHEAD * SEQ * HD;   // 6,291,456 floats
  float* q   = (float*)d_ws;
  float* kb  = q + per;
  float* vb  = kb + per;
  float* att = vb + per;

  // 1) fused QKV projection: [8192,768] x [768,2304]
  gemm64x64_kernel<<<dim3(3 * DIMC / 64, BATCH * SEQ / 64), 128, 0, stream>>>(
      x, w_qkv, 3 * DIMC, /*mode=*/0, q, kb, vb, nullptr, nullptr);

  // 2) flash attention: 96 (b,h) x 16 query blocks
  attn_kernel<<<BATCH * NHEAD * (SEQ / 64), 128, 0, stream>>>(q, kb, vb, att);

  // 3) output projection + bias: [8192,768] x [768,768]
  gemm64x64_kernel<<<dim3(DIMC / 64, BATCH * SEQ / 64), 128, 0, stream>>>(
      att, w_proj, DIMC, /*mode=*/1, nullptr, nullptr, nullptr, b_proj, out);
}